// GCNPredictorNet_12756052869668
// MI455X (gfx1250) — compile-verified
//
#include <hip/hip_runtime.h>
#include <hip/hip_bf16.h>

typedef __attribute__((ext_vector_type(2))) float v2f;
typedef __attribute__((ext_vector_type(8))) float v8f;

#define HID 16
#define NGRAPH 16   // classes
#define NUM_GRAPHS 512

__device__ __forceinline__ void atomAddF(float* p, float v) {
    // relaxed, device-scope fp32 atomic -> global_atomic_add_f32 on gfx1250
    __hip_atomic_fetch_add(p, v, __ATOMIC_RELAXED, __HIP_MEMORY_SCOPE_AGENT);
}

// ---------------- degree / normalization ----------------
__global__ void k_init_deg(float* __restrict__ deg, int n) {
    int i = blockIdx.x * blockDim.x + threadIdx.x;
    if (i < n) deg[i] = 1.0f;   // self-loop
}

__global__ void k_deg_accum(const int* __restrict__ dst, float* __restrict__ deg, int E) {
    int e = blockIdx.x * blockDim.x + threadIdx.x;
    if (e < E) atomAddF(&deg[dst[e]], 1.0f);
}

__global__ void k_rsqrt(float* __restrict__ deg, int n) {
    int i = blockIdx.x * blockDim.x + threadIdx.x;
    if (i < n) deg[i] = rsqrtf(deg[i]);   // deg >= 1 always
}

// ---------------- WMMA GEMM: D[n,16] = (relu?)A[n,K] @ W[K,16] ----------------
// One wave per 16-row tile. fp32 WMMA 16x16x4, K stepped by 4.
template<int K, bool RELU>
__global__ void k_gemm16(const float* __restrict__ A, const float* __restrict__ W,
                         float* __restrict__ D, int n, int n_tiles) {
    const int lane = threadIdx.x & 31;
    const int wid  = threadIdx.x >> 5;
    const int tile = blockIdx.x * (blockDim.x >> 5) + wid;
    if (tile >= n_tiles) return;               // uniform per wave: EXEC stays all-ones

    const int m    = lane & 15;                // row within tile / col of B
    const int half = lane >> 4;                // lane half selects K pair
    const int kh   = half * 2;

    int rowA = tile * 16 + m;
    int rowC = rowA;
    if (rowA >= n) rowA = n - 1;               // clamp loads, keep all lanes active

    const float* arow = A + (size_t)rowA * K;
    v8f acc = {};
#pragma unroll
    for (int k0 = 0; k0 < K; k0 += 4) {
        float a0 = arow[k0 + kh];
        float a1 = arow[k0 + kh + 1];
        if (RELU) { a0 = fmaxf(a0, 0.0f); a1 = fmaxf(a1, 0.0f); }
        v2f a; a.x = a0; a.y = a1;
        v2f b;
        b.x = W[(k0 + kh) * HID + m];
        b.y = W[(k0 + kh + 1) * HID + m];
        acc = __builtin_amdgcn_wmma_f32_16x16x4_f32(false, a, false, b,
                                                    (short)0, acc, false, false);
    }
    // C/D layout: VGPR r -> row (r + 8*half), col m
#pragma unroll
    for (int r = 0; r < 8; ++r) {
        int row = tile * 16 + r + 8 * half;
        if (row < n) D[(size_t)row * HID + m] = acc[r];
    }
}

// ---------------- out = dinv^2 * h + bias (self-loop term) ----------------
__global__ void k_self_bias(const float* __restrict__ h, const float* __restrict__ dinv,
                            const float* __restrict__ bias, float* __restrict__ out, int n) {
    int t = blockIdx.x * blockDim.x + threadIdx.x;
    if (t >= n * HID) return;
    int i = t >> 4, c = t & 15;
    float di = dinv[i];
    out[t] = di * di * h[t] + bias[c];
}

// ---------------- edge scatter: out[d] += dinv[s]*dinv[d]*h[s] ----------------
// 4 lanes per edge, each handles a float4 channel quad.
__global__ void k_edge_prop(const int* __restrict__ src, const int* __restrict__ dst,
                            const float* __restrict__ dinv, const float* __restrict__ h,
                            float* __restrict__ out, int E) {
    int t = blockIdx.x * blockDim.x + threadIdx.x;
    int e = t >> 2;
    int q = t & 3;
    if (e >= E) return;
    int s = src[e], d = dst[e];
    float w = dinv[s] * dinv[d];
    const float4 hv = *(const float4*)(h + (size_t)s * HID + q * 4);
    float* o = out + (size_t)d * HID + q * 4;
    atomAddF(o + 0, w * hv.x);
    atomAddF(o + 1, w * hv.y);
    atomAddF(o + 2, w * hv.z);
    atomAddF(o + 3, w * hv.w);
}

// ---------------- pooling ----------------
__global__ void k_zero_pool(float* __restrict__ sums, float* __restrict__ cnt) {
    int t = blockIdx.x * blockDim.x + threadIdx.x;
    if (t < NUM_GRAPHS * HID) sums[t] = 0.0f;
    if (t < NUM_GRAPHS) cnt[t] = 0.0f;
}

__global__ void k_pool(const float* __restrict__ out, const int* __restrict__ batch,
                       float* __restrict__ sums, float* __restrict__ cnt, int n) {
    int i = blockIdx.x * blockDim.x + threadIdx.x;
    if (i >= n) return;
    int g = batch[i];
    atomAddF(&cnt[g], 1.0f);
    const float* row = out + (size_t)i * HID;
#pragma unroll
    for (int c = 0; c < HID; ++c) atomAddF(&sums[g * HID + c], row[c]);
}

// ---------------- head: y = (sums/max(cnt,1)) @ fcW + fcb ----------------
__global__ void k_head(const float* __restrict__ sums, const float* __restrict__ cnt,
                       const float* __restrict__ fcW, const float* __restrict__ fcb,
                       float* __restrict__ y) {
    int t = blockIdx.x * blockDim.x + threadIdx.x;
    if (t >= NUM_GRAPHS * NGRAPH) return;
    int g = t >> 4, cls = t & 15;
    float inv = 1.0f / fmaxf(cnt[g], 1.0f);
    float acc = fcb[cls];
#pragma unroll
    for (int k = 0; k < HID; ++k)
        acc += (sums[g * HID + k] * inv) * fcW[k * NGRAPH + cls];
    y[t] = acc;
}

extern "C" void kernel_launch(void* const* d_in, const int* in_sizes, int n_in,
                              void* d_out, int out_size, void* d_ws, size_t ws_size,
                              hipStream_t stream) {
    const float* x    = (const float*)d_in[0];
    const float* W1   = (const float*)d_in[1];
    const float* b1   = (const float*)d_in[2];
    const float* W2   = (const float*)d_in[3];
    const float* b2   = (const float*)d_in[4];
    const float* fcW  = (const float*)d_in[5];
    const float* fcb  = (const float*)d_in[6];
    const int*   eidx = (const int*)d_in[7];   // [2, E]
    const int*   batch= (const int*)d_in[8];
    float* y = (float*)d_out;

    const int n = in_sizes[0] / 128;           // IN_CH = 128
    const int E = in_sizes[7] / 2;
    const int* src = eidx;
    const int* dst = eidx + E;

    // workspace layout (16B aligned)
    size_t nAl = ((size_t)n + 3) & ~(size_t)3;
    float* deg  = (float*)d_ws;                     // n (becomes dinv)
    float* h    = deg + nAl;                        // n*16
    float* outb = h + (size_t)n * HID;              // n*16
    float* sums = outb + (size_t)n * HID;           // 512*16
    float* cnt  = sums + NUM_GRAPHS * HID;          // 512

    const int TB = 256;
    const int n_tiles = (n + 15) / 16;

    // normalization
    k_init_deg<<<(n + TB - 1) / TB, TB, 0, stream>>>(deg, n);
    k_deg_accum<<<(E + TB - 1) / TB, TB, 0, stream>>>(dst, deg, E);
    k_rsqrt<<<(n + TB - 1) / TB, TB, 0, stream>>>(deg, n);

    // layer 1: h = x @ W1 (WMMA), out = D^-1/2 (A+I) D^-1/2 h + b1
    k_gemm16<128, false><<<(n_tiles + 3) / 4, 128, 0, stream>>>(x, W1, h, n, n_tiles);
    k_self_bias<<<((n * HID) + TB - 1) / TB, TB, 0, stream>>>(h, deg, b1, outb, n);
    k_edge_prop<<<((E * 4) + TB - 1) / TB, TB, 0, stream>>>(src, dst, deg, h, outb, E);

    // layer 2: h = relu(out) @ W2 (WMMA, fused relu), then propagate + b2
    k_gemm16<16, true><<<(n_tiles + 3) / 4, 128, 0, stream>>>(outb, W2, h, n, n_tiles);
    k_self_bias<<<((n * HID) + TB - 1) / TB, TB, 0, stream>>>(h, deg, b2, outb, n);
    k_edge_prop<<<((E * 4) + TB - 1) / TB, TB, 0, stream>>>(src, dst, deg, h, outb, E);

    // mean pool + classifier head
    k_zero_pool<<<(NUM_GRAPHS * HID + TB - 1) / TB, TB, 0, stream>>>(sums, cnt);
    k_pool<<<(n + TB - 1) / TB, TB, 0, stream>>>(outb, batch, sums, cnt, n);
    k_head<<<(NUM_GRAPHS * NGRAPH + TB - 1) / TB, TB, 0, stream>>>(sums, cnt, fcW, fcb, y);
}